// NormalFloatQuantizer_13537736917200
// MI455X (gfx1250) — compile-verified
//
#include <hip/hip_runtime.h>
#include <stdint.h>

// ---------------- NF4 grid-search quantizer for MI455X (gfx1250) ----------------
// One 256-thread block (8 wave32) per row. Row staged to LDS via the CDNA5
// async data-mover (global_load_async_to_lds_* / ASYNCcnt), then held in VGPRs
// for the 80-factor sweep. Bucketize via a 512-bin LDS LUT (one boundary max
// per bin). All reductions are fixed-order (wave32 shfl tree + per-wave slots)
// so results are bit-identical across replays.

#define COLS   11008
#define GRIDF  80
#define NWAVES 8
#define EPSF   1e-8f
#define NBINS  512
#define RANGE_ 2.0625f                 // covers n in [-1.03125, +1.03125]
#define WBIN   (RANGE_ / 512.0f)       // 33*2^-13, exact in f32
#define INVWF  (512.0f / RANGE_)
#define NMINF  (-1.03125f)             // -33*2^-5, exact; bias = 256.0 exactly

__device__ __forceinline__ int bin_of(float v) {
    float t = fmaf(v, INVWF, 256.0f);  // (v - NMIN) / W
    int j = (int)t;                    // trunc-toward-0; negatives clamp below
    j = j < 0 ? 0 : j;
    j = j > (NBINS - 1) ? (NBINS - 1) : j;
    return j;
}

__global__ __launch_bounds__(256)
void nf4_rowquant_kernel(const float* __restrict__ x,
                         const float* __restrict__ cbg,
                         float* __restrict__ out) {
    __shared__ __align__(16) float  xrow[COLS];            // 44032 B
    __shared__ float4               lut[NBINS];            //  8192 B {bnd, lo, hi, pad}
    __shared__ float                sArr[GRIDF];
    __shared__ float                invArr[GRIDF];
    __shared__ float                errPart[GRIDF][NWAVES];//  2560 B fixed-order partials
    __shared__ float                wred[NWAVES];
    __shared__ float                sBase, sBest;

    const int tid  = threadIdx.x;
    const int lane = tid & 31;
    const int wave = tid >> 5;
    const int row  = blockIdx.x;
    const float* xr = x + (size_t)row * COLS;

    // ---- Phase 0: async-stage the row into LDS (ASYNCcnt path, overlaps setup) ----
    {
        uint32_t ldsBase = (uint32_t)(uintptr_t)(&xrow[0]);
        // 10 sweeps of b128: 256 thr * 16 B = 4096 B each -> 40960 B
        #pragma unroll
        for (int i = 0; i < 10; ++i) {
            uint32_t ldsOff = ldsBase + (uint32_t)(i * 4096 + tid * 16);
            uint64_t ga = (uint64_t)(uintptr_t)((const char*)xr + i * 4096 + tid * 16);
            asm volatile("global_load_async_to_lds_b128 %0, %1, off"
                         :: "v"(ldsOff), "v"(ga) : "memory");
        }
        // remaining 3072 B via 3 b32 sweeps (256 thr * 4 B = 1024 B each)
        #pragma unroll
        for (int i = 0; i < 3; ++i) {
            uint32_t ldsOff = ldsBase + (uint32_t)(40960 + i * 1024 + tid * 4);
            uint64_t ga = (uint64_t)(uintptr_t)((const char*)xr + 40960 + i * 1024 + tid * 4);
            asm volatile("global_load_async_to_lds_b32 %0, %1, off"
                         :: "v"(ldsOff), "v"(ga) : "memory");
        }
    }

    // Codebook + bucket boundaries in registers (uniform across block)
    float cb[16];
    #pragma unroll
    for (int i = 0; i < 16; ++i) cb[i] = cbg[i];
    float bnd[15];
    #pragma unroll
    for (int i = 0; i < 15; ++i) bnd[i] = (cb[i] + cb[i + 1]) * 0.5f;

    // ---- Phase 1: LUT defaults (runs while the async copy is in flight) ----
    #pragma unroll
    for (int r = 0; r < 2; ++r) {
        int   j  = tid + r * 256;
        float lo = fmaf((float)j, WBIN, NMINF);            // exact (33*j * 2^-13)
        int c = 0;
        #pragma unroll
        for (int i = 0; i < 15; ++i) c += (bnd[i] < lo) ? 1 : 0;  // searchsorted 'left'
        lut[j] = make_float4(-1e30f, cb[c], cb[c], 0.0f);
    }
    __syncthreads();
    // Overwrite the (unique) bin of each boundary with a real compare entry.
    if (tid < 15) {
        int j = bin_of(bnd[tid]);
        lut[j] = make_float4(bnd[tid], cb[tid], cb[tid + 1], 0.0f);
    }
    // Each wave waits for its own async copies; the barrier makes it block-wide.
    asm volatile("s_wait_asynccnt 0x0" ::: "memory");
    __syncthreads();

    // ---- Phase 2: row -> VGPRs (stride-256: bank-conflict-free), abs-max ----
    float xv[43];
    float mloc = 0.0f;
    #pragma unroll
    for (int k = 0; k < 43; ++k) {
        xv[k] = xrow[tid + (k << 8)];
        mloc  = fmaxf(mloc, fabsf(xv[k]));
    }
    #pragma unroll
    for (int o = 16; o > 0; o >>= 1)
        mloc = fmaxf(mloc, __shfl_xor(mloc, o, 32));
    if (lane == 0) wred[wave] = mloc;
    __syncthreads();
    if (tid == 0) {
        float b = wred[0];
        #pragma unroll
        for (int w = 1; w < NWAVES; ++w) b = fmaxf(b, wred[w]);
        sBase = fmaxf(b, EPSF);                            // clip(base_scale, EPS)
    }
    __syncthreads();
    if (tid < GRIDF) {
        float fac = (tid == GRIDF - 1) ? 1.25f
                                       : fmaf((float)tid, 0.75f / 79.0f, 0.5f);
        float s   = fmaxf(sBase * fac, EPSF);
        sArr[tid]   = s;
        invArr[tid] = 1.0f / s;
    }
    __syncthreads();

    // ---- Phase 3: 80-factor MSE sweep (x in VGPRs, 1 ds_load_b128 / eval) ----
    for (int f = 0; f < GRIDF; ++f) {
        const float s   = sArr[f];
        const float inv = invArr[f];
        float e = 0.0f;
        #pragma unroll
        for (int k = 0; k < 43; ++k) {
            float  n  = xv[k] * inv;
            int    j  = bin_of(n);
            float4 e4 = lut[j];
            float  q  = (n > e4.x) ? e4.z : e4.y;          // exact searchsorted semantics
            float  d  = fmaf(q, s, -xv[k]);
            e = fmaf(d, d, e);
        }
        #pragma unroll
        for (int o = 16; o > 0; o >>= 1) e += __shfl_xor(e, o, 32);
        if (lane == 0) errPart[f][wave] = e;               // fixed slot: deterministic
    }
    __syncthreads();

    // ---- Phase 4: fixed-order cross-wave sum + sequential first-strict-min ----
    if (tid == 0) {
        float be = INFINITY, bs = sArr[0];
        for (int f = 0; f < GRIDF; ++f) {
            float e = 0.0f;
            #pragma unroll
            for (int w = 0; w < NWAVES; ++w) e += errPart[f][w];  // fixed order
            if (e < be) { be = e; bs = sArr[f]; }          // first strict min wins
        }
        sBest = fmaxf(bs, EPSF);
    }
    __syncthreads();

    // ---- Phase 5: final quantize/dequantize with IEEE division ----
    const float s = sBest;
    float* orow = out + (size_t)row * COLS;
    #pragma unroll
    for (int k = 0; k < 43; ++k) {
        float  n  = xv[k] / s;                             // true div, like reference
        int    j  = bin_of(n);
        float4 e4 = lut[j];
        float  q  = (n > e4.x) ? e4.z : e4.y;
        orow[tid + (k << 8)] = q * s;
    }
}

extern "C" void kernel_launch(void* const* d_in, const int* in_sizes, int n_in,
                              void* d_out, int out_size, void* d_ws, size_t ws_size,
                              hipStream_t stream) {
    (void)n_in; (void)out_size; (void)d_ws; (void)ws_size;
    const float* x  = (const float*)d_in[0];
    const float* cb = (const float*)d_in[1];
    float* out = (float*)d_out;
    int rows = in_sizes[0] / COLS;                         // 4096
    nf4_rowquant_kernel<<<dim3(rows), dim3(256), 0, stream>>>(x, cb, out);
}